// Self_Attn_Downsample_44126493999268
// MI455X (gfx1250) — compile-verified
//
#include <hip/hip_runtime.h>
#include <hip/hip_bf16.h>
#include <stdint.h>

typedef __attribute__((ext_vector_type(2))) float v2f;
typedef __attribute__((ext_vector_type(8))) float v8f;

#define B_    64
#define C_    64
#define N_    1024   // 32*32 spatial after 2x2/s2 conv
#define KPIN_ 1134   // 42*27
#define LOG2E 1.4426950408889634f
#define INV32 0.03125f
#define PAD_  36     // LDS row pad: 16B-aligned rows + conflict-free WMMA reads

// LDS byte offset (wave-relative) of a __shared__ object, for async-to-LDS VDST.
__device__ __forceinline__ uint32_t lds_off(const void* p) {
  return (uint32_t)(uintptr_t)(__attribute__((address_space(3))) const void*)p;
}

// ---------------------------------------------------------------------------
// Kernel 1: q[b,m] = conv2x2(x, wq) + bq   (1 output channel)
// ---------------------------------------------------------------------------
__global__ __launch_bounds__(256) void k_conv_q(const float* __restrict__ x,
                                                const float* __restrict__ wq,
                                                const float* __restrict__ bq,
                                                float* __restrict__ qW) {
  __shared__ float wqS[256];
  int tid = threadIdx.x;
  wqS[tid] = wq[tid];
  __syncthreads();
  int t = blockIdx.x * 256 + tid;          // 65536 total
  int b = t >> 10, m = t & 1023;
  int i = m >> 5, j = m & 31;
  float acc = bq[0];
  const float* xb = x + ((size_t)b << 18); // b*64*64*64
  for (int c = 0; c < 64; ++c) {
    const float* xp = xb + (c << 12) + ((i << 1) << 6) + (j << 1);
    acc += xp[0]  * wqS[c * 4 + 0];
    acc += xp[1]  * wqS[c * 4 + 1];
    acc += xp[64] * wqS[c * 4 + 2];
    acc += xp[65] * wqS[c * 4 + 3];
  }
  qW[t] = acc;
}

// ---------------------------------------------------------------------------
// Kernel 2: kfeat[b,p] = conv2x2(v, wk) + bk   (1 output channel, 42x27)
// ---------------------------------------------------------------------------
__global__ __launch_bounds__(256) void k_conv_k(const float* __restrict__ v,
                                                const float* __restrict__ wk,
                                                const float* __restrict__ bk,
                                                float* __restrict__ kfW) {
  __shared__ float wkS[256];
  int tid = threadIdx.x;
  wkS[tid] = wk[tid];
  __syncthreads();
  int t = blockIdx.x * 256 + tid;
  if (t >= B_ * KPIN_) return;
  int b = t / KPIN_, p = t % KPIN_;
  int i = p / 27, j = p % 27;
  float acc = bk[0];
  const float* vb = v + (size_t)b * 64 * 84 * 54;
  for (int c = 0; c < 64; ++c) {
    const float* vp = vb + c * 84 * 54 + (i * 2) * 54 + (j * 2);
    acc += vp[0]  * wkS[c * 4 + 0];
    acc += vp[1]  * wkS[c * 4 + 1];
    acc += vp[54] * wkS[c * 4 + 2];
    acc += vp[55] * wkS[c * 4 + 3];
  }
  kfW[t] = acc;
}

// ---------------------------------------------------------------------------
// Kernel 3: k[b,n] = kfeat[b,:] @ wproj[n,:]^T + bproj[n]  (one block / batch)
// ---------------------------------------------------------------------------
__global__ __launch_bounds__(256) void k_kproj(const float* __restrict__ kfW,
                                               const float* __restrict__ wproj,
                                               const float* __restrict__ bproj,
                                               float* __restrict__ kW) {
  __shared__ float kfS[KPIN_];
  int b = blockIdx.x, tid = threadIdx.x;
  for (int idx = tid; idx < KPIN_; idx += 256) kfS[idx] = kfW[b * KPIN_ + idx];
  __syncthreads();
  for (int rep = 0; rep < 4; ++rep) {
    int n = tid + rep * 256;
    float acc = bproj[n];
    const float* wrow = wproj + (size_t)n * KPIN_;
    for (int j = 0; j < KPIN_; ++j) acc += kfS[j] * wrow[j];
    kW[(b << 10) + n] = acc;
  }
}

// ---------------------------------------------------------------------------
// Kernel 4: pv[b,c,n] = conv2x2(x, wv) + bv  (block = one batch, one i-row)
// ---------------------------------------------------------------------------
__global__ __launch_bounds__(256) void k_conv_pv(const float* __restrict__ x,
                                                 const float* __restrict__ wv,
                                                 const float* __restrict__ bv,
                                                 float* __restrict__ pvW) {
  __shared__ float xT[64 * 2 * 64];  // [c2][h2][w] = 32 KB
  int i   = blockIdx.x;              // 0..31 output row
  int b   = blockIdx.y;
  int tid = threadIdx.x;
  const float* xb = x + ((size_t)b << 18);
  for (int idx = tid; idx < 8192; idx += 256) {
    int c2 = idx >> 7, rem = idx & 127, h2 = rem >> 6, ww = rem & 63;
    xT[idx] = xb[(c2 << 12) + ((i * 2 + h2) << 6) + ww];
  }
  __syncthreads();
  int j = tid & 31, cg = tid >> 5;  // cg in 0..7
  for (int cc = 0; cc < 8; ++cc) {
    int c = cg * 8 + cc;
    float acc = bv[c];
    const float* wr = wv + (c << 8);   // uniform per wave -> scalar loads
    for (int c2 = 0; c2 < 64; ++c2) {
      const float* xp = &xT[c2 * 128 + (j << 1)];
      acc += xp[0]  * wr[c2 * 4 + 0];
      acc += xp[1]  * wr[c2 * 4 + 1];
      acc += xp[64] * wr[c2 * 4 + 2];
      acc += xp[65] * wr[c2 * 4 + 3];
    }
    pvW[(((size_t)b * 64 + c) << 10) + (i << 5) + j] = acc;
  }
}

// ---------------------------------------------------------------------------
// Kernel 5 (fused): rank-1 softmax -> attn store + f32 WMMA GEMM (pv @ attn^T)
// with double-buffered ASYNC global->LDS staging, then 2x2 transposed conv.
// grid = (64 m-tiles, 64 batches), block = 128 (4 waves; wave w owns c-tile w)
// ---------------------------------------------------------------------------
__global__ __launch_bounds__(128) void k_fused_attn(
    const float* __restrict__ qW, const float* __restrict__ kW,
    const float* __restrict__ pvW, const float* __restrict__ wu,
    const float* __restrict__ bu, float* __restrict__ upOut,
    float* __restrict__ attnOut) {
  __shared__ float kS[1024];
  __shared__ float qS[16], qsS[16], rmLS[16], invZS[16];
  __shared__ float partS[128], minS[128], kmmS[2];
  __shared__ __align__(16) float pvT[2][64 * PAD_];   // [buf][c][n]
  __shared__ __align__(16) float attnT[2][16 * PAD_]; // [buf][m][n]
  __shared__ float outF[64 * 17];                     // [c][m] padded

  const int mt = blockIdx.x, b = blockIdx.y;
  const int tid = threadIdx.x, lane = tid & 31, w = tid >> 5;

  // --- stage k row, track local max/min ---
  float lmax = -3.4e38f, lmin = 3.4e38f;
  for (int r = 0; r < 8; ++r) {
    int n = tid + (r << 7);
    float kv = kW[(b << 10) + n];
    kS[n] = kv;
    lmax = fmaxf(lmax, kv);
    lmin = fminf(lmin, kv);
  }
  partS[tid] = lmax;
  minS[tid] = lmin;
  if (tid < 16) qS[tid] = qW[(b << 10) + (mt << 4) + tid];
  __syncthreads();
  if (tid < 16) {  // two-level reduce: 128 -> 16
    float mx = partS[tid], mn = minS[tid];
    for (int t = tid + 16; t < 128; t += 16) {
      mx = fmaxf(mx, partS[t]);
      mn = fminf(mn, minS[t]);
    }
    partS[tid] = mx;
    minS[tid] = mn;
  }
  __syncthreads();
  if (tid == 0) {  // 16 -> 1
    float mx = partS[0], mn = minS[0];
    for (int t = 1; t < 16; ++t) {
      mx = fmaxf(mx, partS[t]);
      mn = fminf(mn, minS[t]);
    }
    kmmS[0] = mx;
    kmmS[1] = mn;
  }
  __syncthreads();
  // per-row log2-domain scale and analytic rowmax (rank-1 energy)
  if (tid < 16) {
    float qm = qS[tid];
    float qs = qm * (INV32 * LOG2E);
    qsS[tid] = qs;
    rmLS[tid] = qs * (qm >= 0.0f ? kmmS[0] : kmmS[1]);
  }
  __syncthreads();
  // row sums Z
  {
    int mi = tid >> 3, seg = tid & 7;
    float qs = qsS[mi], rm = rmLS[mi], s = 0.0f;
    for (int nn = seg * 128; nn < seg * 128 + 128; ++nn)
      s += exp2f(qs * kS[nn] - rm);
    partS[tid] = s;
  }
  __syncthreads();
  if (tid < 16) {
    float s = 0.0f;
    for (int t = 0; t < 8; ++t) s += partS[tid * 8 + t];
    invZS[tid] = 1.0f / s;
  }
  __syncthreads();

  const float* pvB = pvW + (((size_t)b) << 16);  // b*64*1024
  float* attnB = attnOut + ((((size_t)b << 10) + (mt << 4)) << 10);

  // issue async DMA of pv tile [64c x 32n] for `chunk` into buffer `buf`
  auto issue_pv = [&](int chunk, int buf) {
    int n0 = chunk << 5;
    for (int idx = tid; idx < 512; idx += 128) {  // 512 x B128 quads
      int c = idx >> 3, q4 = (idx & 7) << 2;
      uint32_t ldst = lds_off(&pvT[buf][c * PAD_ + q4]);
      uint64_t gsrc = (uint64_t)(uintptr_t)(pvB + (c << 10) + n0 + q4);
      asm volatile("global_load_async_to_lds_b128 %0, %1, off"
                   :: "v"(ldst), "v"(gsrc) : "memory");
    }
  };
  // compute normalized attn tile [16m x 32n] into `buf`, also store to global
  auto gen_attn = [&](int chunk, int buf) {
    int n0 = chunk << 5;
    for (int idx = tid; idx < 512; idx += 128) {
      int mi = idx >> 5, noff = idx & 31;
      float val = exp2f(qsS[mi] * kS[n0 + noff] - rmLS[mi]) * invZS[mi];
      attnT[buf][mi * PAD_ + noff] = val;
      attnB[(mi << 10) + n0 + noff] = val;
    }
  };

  // --- main loop: 32 chunks of 32 n, double-buffered ---
  v8f acc = {};
  issue_pv(0, 0);
  gen_attn(0, 0);
  asm volatile("s_wait_asynccnt 0x0" ::: "memory");
  __syncthreads();

  const int row = lane & 15;          // A: c within tile / B: m column
  const int koff = (lane >> 4) << 1;  // lanes 16-31 hold K=2,3
  for (int chunk = 0; chunk < 32; ++chunk) {
    int buf = chunk & 1;
    if (chunk + 1 < 32) {             // prefetch next chunk into other buffer
      issue_pv(chunk + 1, buf ^ 1);
      gen_attn(chunk + 1, buf ^ 1);
    }
    // WMMA: D[16c x 16m] += A(pv 16x4) x B(attn^T 4x16), 8 k-steps of 4
    {
      const float* aRow = &pvT[buf][(w * 16 + row) * PAD_];
      const float* bRow = &attnT[buf][row * PAD_];
      for (int ks = 0; ks < 8; ++ks) {
        int ni = (ks << 2) + koff;
        v2f a, bb;
        a.x = aRow[ni];
        a.y = aRow[ni + 1];
        bb.x = bRow[ni];
        bb.y = bRow[ni + 1];
        acc = __builtin_amdgcn_wmma_f32_16x16x4_f32(
            false, a, false, bb, (short)0, acc, false, false);
      }
    }
    asm volatile("s_wait_asynccnt 0x0" ::: "memory");
    __syncthreads();
  }

  // spill C layout: VGPR r, lane<16 -> M=r ; lane>=16 -> M=8+r ; N=lane&15
  {
    int mcol = lane & 15;
    int radd = (lane & 16) ? 8 : 0;
    for (int r = 0; r < 8; ++r)
      outF[(w * 16 + r + radd) * 17 + mcol] = acc[r];
  }
  __syncthreads();

  // --- transposed conv 2x2/s2: up[b,d,2i+a,2j+b2] = sum_c out[c,m]*wu[c,d,a,b2]+bu
  {
    int g = tid >> 5;          // d-group 0..3
    int woff = tid & 31;       // output w offset (2*mi + b2)
    int mi2 = woff >> 1, b2 = woff & 1;
    int i = mt >> 1;           // m-tile -> spatial row
    int j0 = (mt & 1) << 4;    // 0 or 16
    float* upB = upOut + (((size_t)b) << 18);
    for (int dd = 0; dd < 16; ++dd) {
      int d = (g << 4) + dd;
      const float* wuD = wu + (d << 2);
      for (int a = 0; a < 2; ++a) {
        float acc2 = bu[d];
        int wsel = (a << 1) + b2;
        for (int c = 0; c < 64; ++c)
          acc2 += outF[c * 17 + mi2] * wuD[(c << 8) + wsel];
        upB[(d << 12) + (((i << 1) + a) << 6) + (j0 << 1) + woff] = acc2;
      }
    }
  }
}

// ---------------------------------------------------------------------------
extern "C" void kernel_launch(void* const* d_in, const int* in_sizes, int n_in,
                              void* d_out, int out_size, void* d_ws,
                              size_t ws_size, hipStream_t stream) {
  const float* x     = (const float*)d_in[0];
  const float* v     = (const float*)d_in[1];
  const float* wq    = (const float*)d_in[2];
  const float* bq    = (const float*)d_in[3];
  const float* wk    = (const float*)d_in[4];
  const float* bk    = (const float*)d_in[5];
  const float* wv    = (const float*)d_in[6];
  const float* bv    = (const float*)d_in[7];
  const float* wu    = (const float*)d_in[8];
  const float* bu    = (const float*)d_in[9];
  const float* wproj = (const float*)d_in[10];
  const float* bproj = (const float*)d_in[11];

  float* up   = (float*)d_out;                 // 64*64*64*64 = 16777216
  float* attn = up + 16777216;                 // 64*1024*1024 = 67108864

  float* ws  = (float*)d_ws;
  float* qW  = ws;                  // 65536
  float* kfW = qW + 65536;          // 72576
  float* kW  = kfW + 72576;         // 65536
  float* pvW = kW + 65536;          // 4194304  (~17.6 MB total)

  k_conv_q<<<256, 256, 0, stream>>>(x, wq, bq, qW);
  k_conv_k<<<(B_ * KPIN_ + 255) / 256, 256, 0, stream>>>(v, wk, bk, kfW);
  k_kproj<<<64, 256, 0, stream>>>(kfW, wproj, bproj, kW);
  k_conv_pv<<<dim3(32, 64), 256, 0, stream>>>(x, wv, bv, pvW);
  k_fused_attn<<<dim3(64, 64), 128, 0, stream>>>(qW, kW, pvW, wu, bu, up, attn);
}